// SAGE_Net_34093450396002
// MI455X (gfx1250) — compile-verified
//
#include <hip/hip_runtime.h>

typedef float v2f __attribute__((ext_vector_type(2)));
typedef float v8f __attribute__((ext_vector_type(8)));

#define TK 128        // K dimension (both layers have K=128)
#define LDS_STRIDE (TK + 4)  // pad: row stride 132 words -> conflict-free A reads

// ---------------------------------------------------------------- utilities
__global__ void zero_kernel(float* __restrict__ p, long long n) {
  long long i = (long long)blockIdx.x * blockDim.x + threadIdx.x;
  if (i < n) p[i] = 0.0f;
}

__global__ void degree_kernel(const int* __restrict__ dst, float* __restrict__ deg, int E) {
  int e = blockIdx.x * blockDim.x + threadIdx.x;
  if (e < E) atomicAdd(&deg[dst[e]], 1.0f);
}

// ------------------------------------------------- dual-weight WMMA GEMM
// Computes outl = X @ Wl^T, outr = X @ Wr^T for X:[nrows,128], W:[hout,128].
// One wave owns a 16-row strip; A fragments held in registers across all
// output tiles of both weight matrices.
__global__ __launch_bounds__(128) void gemm_dual_wmma(
    const float* __restrict__ X,
    const float* __restrict__ Wl, const float* __restrict__ Wr,
    float* __restrict__ outl, float* __restrict__ outr,
    int nrows, int hout)
{
  __shared__ float lds[4][16 * LDS_STRIDE];   // 4 waves * ~8.25KB = 33KB
  const int wave = threadIdx.x >> 5;
  const int lane = threadIdx.x & 31;
  const int row0 = (blockIdx.x * 4 + wave) * 16;

  // coalesced stage of this wave's 16x128 A-strip into LDS (padded rows)
  float* myLds = lds[wave];
#pragma unroll 4
  for (int i = 0; i < 64; ++i) {
    int idx = i * 32 + lane;                 // 0..2047
    int r   = idx >> 7;
    int c   = idx & (TK - 1);
    int gr  = row0 + r;
    myLds[r * LDS_STRIDE + c] = (gr < nrows) ? X[(size_t)gr * TK + c] : 0.0f;
  }
  __syncthreads();

  const int m  = lane & 15;   // row within tile (A), col within tile (B/D)
  const int kh = lane >> 4;   // K-half select for 16x16x4 f32 layout

  // A fragments: lane holds A[m][kb*4 + kh*2 + {0,1}]
  v2f a[TK / 4];
#pragma unroll
  for (int kb = 0; kb < TK / 4; ++kb) {
    int k = kb * 4 + kh * 2;
    a[kb][0] = myLds[m * LDS_STRIDE + k];
    a[kb][1] = myLds[m * LDS_STRIDE + k + 1];
  }

  const int ntiles = hout >> 4;
  for (int t = 0; t < 2 * ntiles; ++t) {
    const float* W  = (t < ntiles) ? Wl : Wr;
    float* outp     = (t < ntiles) ? outl : outr;
    const int nb    = ((t < ntiles) ? t : t - ntiles) * 16;
    const float* wrow = W + (size_t)(nb + m) * TK;  // B[k][n] = W[nb+n][k]
    v8f acc = {};
#pragma unroll
    for (int kb = 0; kb < TK / 4; ++kb) {
      int k = kb * 4 + kh * 2;
      v2f b;
      b[0] = wrow[k];
      b[1] = wrow[k + 1];
      acc = __builtin_amdgcn_wmma_f32_16x16x4_f32(
          false, a[kb], false, b, (short)0, acc, false, false);
    }
    // D layout: VGPR v -> row (v + kh*8), col (nb + m)
#pragma unroll
    for (int v = 0; v < 8; ++v) {
      int gr = row0 + v + kh * 8;
      if (gr < nrows) outp[(size_t)gr * hout + nb + m] = acc[v];
    }
  }
}

// ------------------------------------------- per-edge gather + scatter-add
// C/4 threads per edge; each thread moves a float4 (coalesced 512B per edge
// for C=128 => one wave per edge).
template <int C>
__global__ void scatter_add_kernel(const float* __restrict__ feat,
                                   const int* __restrict__ src,
                                   const int* __restrict__ dst,
                                   float* __restrict__ agg, int E)
{
  const int per = C / 4;
  long long tid = (long long)blockIdx.x * blockDim.x + threadIdx.x;
  if (tid >= (long long)E * per) return;
  int e = (int)(tid / per);
  int c = (int)(tid % per) * 4;
  int s = src[e];
  int d = dst[e];
  const float4 v = *(const float4*)(feat + (size_t)s * C + c);
  float* p = agg + (size_t)d * C + c;
  atomicAdd(p + 0, v.x);
  atomicAdd(p + 1, v.y);
  atomicAdd(p + 2, v.z);
  atomicAdd(p + 3, v.w);
}

// out = (agg / max(deg,1)) + self + bias [+ relu]
__global__ void finalize_kernel(const float* __restrict__ agg,
                                const float* __restrict__ self,
                                const float* __restrict__ bias,
                                const float* __restrict__ deg,
                                float* __restrict__ out,
                                long long total, int cshift, int do_relu)
{
  long long tid = (long long)blockIdx.x * blockDim.x + threadIdx.x;
  if (tid >= total) return;
  int r = (int)(tid >> cshift);
  int c = (int)(tid & ((1 << cshift) - 1));
  float d = deg[r];
  d = d > 0.0f ? d : 1.0f;
  float val = agg[tid] / d + self[tid] + bias[c];
  if (do_relu) val = fmaxf(val, 0.0f);
  out[tid] = val;
}

// ---------------------------------------------------------------- launcher
extern "C" void kernel_launch(void* const* d_in, const int* in_sizes, int n_in,
                              void* d_out, int out_size, void* d_ws, size_t ws_size,
                              hipStream_t stream)
{
  const float* x   = (const float*)d_in[0];
  const float* W1l = (const float*)d_in[1];
  const float* W1r = (const float*)d_in[2];
  const float* b1  = (const float*)d_in[3];
  const float* W2l = (const float*)d_in[4];
  const float* W2r = (const float*)d_in[5];
  const float* b2  = (const float*)d_in[6];
  const int*   ei  = (const int*)d_in[7];

  const int IN_ = 128, H_ = 128, OUT_ = 64;
  const int N = in_sizes[0] / IN_;
  const int E = in_sizes[7] / 2;
  const int* srcp = ei;       // edge_index row 0
  const int* dstp = ei + E;   // edge_index row 1

  // workspace layout (floats): [deg N][agg N*128][bufA N*128][bufB N*128]
  float* ws   = (float*)d_ws;
  float* deg  = ws;
  float* agg  = deg + N;                  // agg1; later hl | hr
  float* bufA = agg  + (size_t)N * H_;    // xl;   later agg2
  float* bufB = bufA + (size_t)N * H_;    // xr -> h (in place)
  float* out  = (float*)d_out;

  // 1) zero deg + agg1 (contiguous region)
  {
    long long cnt = (long long)N * (H_ + 1);
    zero_kernel<<<(int)((cnt + 255) / 256), 256, 0, stream>>>(deg, cnt);
  }
  // 2) degrees
  degree_kernel<<<(E + 255) / 256, 256, 0, stream>>>(dstp, deg, E);
  // 3) xl = x@W1l^T -> bufA ; xr = x@W1r^T -> bufB   (WMMA)
  gemm_dual_wmma<<<(N + 63) / 64, 128, 0, stream>>>(x, W1l, W1r, bufA, bufB, N, H_);
  // 4) agg1 += xl[src] at dst (128-wide)
  {
    long long total = (long long)E * (H_ / 4);
    scatter_add_kernel<128><<<(int)((total + 255) / 256), 256, 0, stream>>>(
        bufA, srcp, dstp, agg, E);
  }
  // 5) h = relu(agg1/deg + xr + b1) -> bufB (in place)
  {
    long long total = (long long)N * H_;
    finalize_kernel<<<(int)((total + 255) / 256), 256, 0, stream>>>(
        agg, bufB, b1, deg, bufB, total, 7, 1);
  }
  // 6) zero agg2 (bufA reused, first N*64)
  {
    long long cnt = (long long)N * OUT_;
    zero_kernel<<<(int)((cnt + 255) / 256), 256, 0, stream>>>(bufA, cnt);
  }
  // 7) hl = h@W2l^T -> agg[0..N*64) ; hr = h@W2r^T -> agg[N*64..)   (WMMA)
  gemm_dual_wmma<<<(N + 63) / 64, 128, 0, stream>>>(
      bufB, W2l, W2r, agg, agg + (size_t)N * OUT_, N, OUT_);
  // 8) agg2 += hl[src] at dst (64-wide — halved edge traffic vs aggregating h)
  {
    long long total = (long long)E * (OUT_ / 4);
    scatter_add_kernel<64><<<(int)((total + 255) / 256), 256, 0, stream>>>(
        agg, srcp, dstp, bufA, E);
  }
  // 9) out = agg2/deg + hr + b2
  {
    long long total = (long long)N * OUT_;
    finalize_kernel<<<(int)((total + 255) / 256), 256, 0, stream>>>(
        bufA, agg + (size_t)N * OUT_, b2, deg, out, total, 6, 0);
  }
}